// Value_Predictor_29008209117366
// MI455X (gfx1250) — compile-verified
//
#include <hip/hip_runtime.h>
#include <math.h>

// ---------------------------------------------------------------------------
// GNN forward: 2x GraphConv(128) + single-node classifier head.
// N=50000 (== 3125*16), E=800000, feature dim 128.
// ---------------------------------------------------------------------------

typedef __attribute__((ext_vector_type(2))) float v2f;
typedef __attribute__((ext_vector_type(8))) float v8f;

__global__ void k_zero(float* __restrict__ p, long n) {
    long i = (long)blockIdx.x * blockDim.x + threadIdx.x;
    if (i < n) p[i] = 0.0f;
}

// atomic degree count: deg_out[src[e]] += 1 ; deg_in[dst[e]] += 1
__global__ void k_deg(const int* __restrict__ src, const int* __restrict__ dst,
                      float* __restrict__ deg_out, float* __restrict__ deg_in, int E) {
    int e = blockIdx.x * blockDim.x + threadIdx.x;
    if (e < E) {
        atomicAdd(&deg_out[src[e]], 1.0f);
        atomicAdd(&deg_in[dst[e]], 1.0f);
    }
}

// in place: p[i] = max(p[i],1)^-0.5
__global__ void k_norm(float* __restrict__ p, long n) {
    long i = (long)blockIdx.x * blockDim.x + threadIdx.x;
    if (i < n) p[i] = 1.0f / sqrtf(fmaxf(p[i], 1.0f));
}

// Y[N,128] = (X * rs[:,None]) @ W   with W[128,128], via V_WMMA_F32_16X16X4_F32.
// grid.x = N/16 row tiles; blockDim = 256 (8 waves); wave w handles column tile w.
__global__ void k_gemm_rowscale(const float* __restrict__ X, const float* __restrict__ rs,
                                const float* __restrict__ W, float* __restrict__ Y) {
    const int lane   = threadIdx.x & 31;
    const int tile_n = threadIdx.x >> 5;   // 0..7 -> 16-col tile
    const int tile_m = blockIdx.x;         // 16-row tile
    const int half   = lane >> 4;          // 0 or 1
    const int l16    = lane & 15;

    const int row = tile_m * 16 + l16;     // A row for this lane
    const int col = tile_n * 16 + l16;     // B/C column for this lane
    const float scale = rs[row];
    const float* __restrict__ xrow = X + (long)row * 128;

    v8f acc = {};
    #pragma unroll
    for (int k = 0; k < 128; k += 4) {
        // A fragment (16x4): VGPR0 = K = k+2*half, VGPR1 = K = k+1+2*half
        v2f a, b;
        a.x = xrow[k + 2 * half]     * scale;
        a.y = xrow[k + 2 * half + 1] * scale;
        // B fragment (4x16): same K pairing, striped over 16 lanes per row
        b.x = W[(k + 2 * half)     * 128 + col];
        b.y = W[(k + 2 * half + 1) * 128 + col];
        acc = __builtin_amdgcn_wmma_f32_16x16x4_f32(
            /*neg_a=*/false, a, /*neg_b=*/false, b,
            /*c_mod=*/(short)0, acc, /*reuse_a=*/false, /*reuse_b=*/false);
    }
    // C layout: VGPR j -> row (j + 8*half), col l16 within the 16x16 tile
    float* __restrict__ ybase = Y + (long)(tile_m * 16) * 128 + tile_n * 16;
    #pragma unroll
    for (int j = 0; j < 8; ++j)
        ybase[(long)(j + 8 * half) * 128 + l16] = acc[j];
}

// agg[dst[e], :] += m[src[e], :]  -- one thread per (edge, 4-float chunk)
__global__ void k_scatter(const float* __restrict__ m, const int* __restrict__ src,
                          const int* __restrict__ dst, float* __restrict__ agg, int E) {
    long tid = (long)blockIdx.x * blockDim.x + threadIdx.x;
    if (tid >= (long)E * 32) return;
    int e = (int)(tid >> 5);
    int c = (int)(tid & 31) << 2;
    int s = src[e], d = dst[e];
    const float4 v = *(const float4*)(m + (long)s * 128 + c);
    float* out = agg + (long)d * 128 + c;
    atomicAdd(out + 0, v.x);
    atomicAdd(out + 1, v.y);
    atomicAdd(out + 2, v.z);
    atomicAdd(out + 3, v.w);
}

// h[i] = lrelu(agg[i] * ndst[i/128] + bias[i%128])
__global__ void k_act(const float* __restrict__ agg, const float* __restrict__ ndst,
                      const float* __restrict__ bias, float* __restrict__ h, long n) {
    long i = (long)blockIdx.x * blockDim.x + threadIdx.x;
    if (i >= n) return;
    float v = agg[i] * ndst[i >> 7] + bias[i & 127];
    h[i] = v >= 0.0f ? v : 0.01f * v;
}

// Classifier head evaluated only at node_index; ~2k FLOPs -> single thread.
__global__ void k_final(const float* __restrict__ h2,
                        const float* __restrict__ Wr, const float* __restrict__ br,
                        const float* __restrict__ Wp, const float* __restrict__ bp,
                        const int* __restrict__ node_index, const int* __restrict__ label,
                        float* __restrict__ out) {
    if (threadIdx.x != 0 || blockIdx.x != 0) return;
    const int ni = node_index[0];
    const int lb = label[0];
    const float* t = h2 + (long)ni * 128;

    // logits_raw = t @ Wr + br  (Wr is [128,7])
    float lr[7];
    for (int c = 0; c < 7; ++c) {
        float s = br[c];
        for (int k = 0; k < 128; ++k) s += t[k] * Wr[k * 7 + c];
        lr[c] = s;
    }
    float mx = lr[0];
    for (int c = 1; c < 7; ++c) mx = fmaxf(mx, lr[c]);
    float se = 0.0f;
    for (int c = 0; c < 7; ++c) se += expf(lr[c] - mx);
    float lse = mx + logf(se);
    float lg[7];
    for (int c = 0; c < 7; ++c) lg[c] = lr[c] - lse;

    // hv = concat(t, lg) @ Wp + bp  (Wp is [135,7])
    float hv[7];
    for (int c = 0; c < 7; ++c) {
        float s = bp[c];
        for (int k = 0; k < 128; ++k) s += t[k] * Wp[k * 7 + c];
        for (int k = 0; k < 7; ++k)   s += lg[k] * Wp[(128 + k) * 7 + c];
        hv[c] = s;
    }
    float mx2 = hv[0];
    for (int c = 1; c < 7; ++c) mx2 = fmaxf(mx2, hv[c]);
    float se2 = 0.0f;
    for (int c = 0; c < 7; ++c) se2 += expf(hv[c] - mx2);
    float lse2 = mx2 + logf(se2);
    out[0] = -(hv[lb] - lse2);
}

extern "C" void kernel_launch(void* const* d_in, const int* in_sizes, int n_in,
                              void* d_out, int out_size, void* d_ws, size_t ws_size,
                              hipStream_t stream) {
    const float* feat = (const float*)d_in[0];
    const float* W1   = (const float*)d_in[1];
    const float* b1   = (const float*)d_in[2];
    const float* W2   = (const float*)d_in[3];
    const float* b2   = (const float*)d_in[4];
    const float* Wr   = (const float*)d_in[5];
    const float* br   = (const float*)d_in[6];
    const float* Wp   = (const float*)d_in[7];
    const float* bp   = (const float*)d_in[8];
    const int*   src  = (const int*)d_in[9];
    const int*   dst  = (const int*)d_in[10];
    const int*   nidx = (const int*)d_in[11];
    const int*   labl = (const int*)d_in[12];

    const int  N  = in_sizes[0] / 128;   // 50000, multiple of 16
    const int  E  = in_sizes[9];         // 800000
    const long NF = (long)N * 128;

    // workspace layout (floats): nsrc[N] | ndst[N] | m[N*128] | agg[N*128] | h[N*128]
    float* ws   = (float*)d_ws;
    float* nsrc = ws;                 // also used as deg_out during counting
    float* ndst = ws + N;             // also used as deg_in  during counting
    float* m    = ws + 2L * N;
    float* agg  = m + NF;
    float* h    = agg + NF;
    if (ws_size < (size_t)(2L * N + 3L * NF) * sizeof(float)) return;

    const int T = 256;
    const int gN2  = (int)((2L * N + T - 1) / T);
    const int gE   = (E + T - 1) / T;
    const int gNF  = (int)((NF + T - 1) / T);
    const int gSC  = (int)(((long)E * 32 + T - 1) / T);
    const int gMM  = N / 16;

    // degrees -> normalizers
    k_zero<<<gN2, T, 0, stream>>>(nsrc, 2L * N);
    k_deg <<<gE,  T, 0, stream>>>(src, dst, nsrc, ndst, E);
    k_norm<<<gN2, T, 0, stream>>>(nsrc, 2L * N);

    // layer 1
    k_gemm_rowscale<<<gMM, T, 0, stream>>>(feat, nsrc, W1, m);
    k_zero   <<<gNF, T, 0, stream>>>(agg, NF);
    k_scatter<<<gSC, T, 0, stream>>>(m, src, dst, agg, E);
    k_act    <<<gNF, T, 0, stream>>>(agg, ndst, b1, h, NF);

    // layer 2 (reuse buffers)
    k_gemm_rowscale<<<gMM, T, 0, stream>>>(h, nsrc, W2, m);
    k_zero   <<<gNF, T, 0, stream>>>(agg, NF);
    k_scatter<<<gSC, T, 0, stream>>>(m, src, dst, agg, E);
    k_act    <<<gNF, T, 0, stream>>>(agg, ndst, b2, h, NF);

    // single-node head + loss
    k_final<<<1, 32, 0, stream>>>(h, Wr, br, Wp, bp, nidx, labl, (float*)d_out);
}